// Net_SDE_61881888801193
// MI455X (gfx1250) — compile-verified
//
#include <hip/hip_runtime.h>
#include <hip/hip_bf16.h>

typedef float v2f __attribute__((ext_vector_type(2)));
typedef float v8f __attribute__((ext_vector_type(8)));

#define N_STEPS 48
#define MC_PATHS 131072
#define NKC 13                 // K chunks of 4 (52 >= 50)
#define NNT 4                  // N tiles of 16 (64 >= 50)
#define WAVES_PER_BLOCK 4
#define BLOCK_T (WAVES_PER_BLOCK * 32)
#define PATHS_PER_BLOCK (WAVES_PER_BLOCK * 16)
#define H1_STRIDE 68           // padded row stride (floats) -> bank-conflict-free
#define H_STEP (2.0f / 48.0f)

struct MlpW { const float *W1, *W2, *b1, *b2, *Wo; };

__device__ __forceinline__ float softplusf(float x) {
    // jax.nn.softplus = max(x,0) + log1p(exp(-|x|))
    return fmaxf(x, 0.f) + log1pf(expf(-fabsf(x)));
}

// ---------------- prep: zero accumulators + rho(t) table -------------------
__global__ void prep_kernel(MlpW mRho, float* gAcc, float* rhoTab, float* sqTab) {
    int i = threadIdx.x;
    if (i < 24 * 40) gAcc[i] = 0.f;
    if (i < N_STEPS) {
        float t = i * H_STEP;
        float h1[5], h2[5];
        #pragma unroll
        for (int j = 0; j < 5; ++j) h1[j] = fmaxf(mRho.W1[j] * t + mRho.b1[j], 0.f);
        #pragma unroll
        for (int j = 0; j < 5; ++j) {
            float a = mRho.b2[j];
            #pragma unroll
            for (int k = 0; k < 5; ++k) a = fmaf(mRho.W2[j * 5 + k], h1[k], a);
            h2[j] = fmaxf(a, 0.f);
        }
        float o = 0.f;
        #pragma unroll
        for (int k = 0; k < 5; ++k) o = fmaf(mRho.Wo[k], h2[k], o);
        float rho = tanhf(o);
        rhoTab[i] = rho;
        sqTab[i]  = sqrtf(fmaxf(1.f - rho * rho, 0.f));
    }
}

// ---------------- main SDE simulation kernel -------------------------------
__global__ __launch_bounds__(BLOCK_T) void sde_kernel(
    MlpW mS, MlpW mDr, MlpW mDv,
    const float* __restrict__ S0p, const float* __restrict__ V0p,
    const float* __restrict__ ratep,
    const float* __restrict__ z, const float* __restrict__ z1,
    const float* __restrict__ rhoTab, const float* __restrict__ sqTab,
    float* __restrict__ gAcc)
{
    __shared__ __align__(16) v2f   sB[3 * NKC * NNT * 32];          // W2^T in B-reg order
    __shared__ __align__(16) float sH[WAVES_PER_BLOCK][16 * H1_STRIDE]; // per-wave h1 tile
    __shared__ float sAcc[24 * 40];

    const int tid  = threadIdx.x;
    const int lane = tid & 31;
    const int wv   = tid >> 5;
    const int l16  = lane & 15;
    const int hi   = lane >> 4;

    for (int i = tid; i < 24 * 40; i += BLOCK_T) sAcc[i] = 0.f;

    // --- fill B tables: sB[((m*NKC+kc)*NNT+nt)*32 + lane] = (W2[n][k], W2[n][k+1])
    {
        const float* W2s[3] = { mS.W2, mDr.W2, mDv.W2 };
        for (int i = tid; i < 3 * NKC * NNT * 32; i += BLOCK_T) {
            int ln = i & 31;
            int nt = (i >> 5) & 3;
            int rest = i >> 7;
            int kc = rest % NKC;
            int mp = rest / NKC;
            int n = nt * 16 + (ln & 15);
            int k = 4 * kc + 2 * (ln >> 4);
            const float* W2 = W2s[mp];
            float f0 = (n < 50 && k     < 50) ? W2[n * 50 + k]     : 0.f;
            float f1 = (n < 50 && k + 1 < 50) ? W2[n * 50 + k + 1] : 0.f;
            sB[i] = (v2f){ f0, f1 };
        }
    }
    __syncthreads();

    // --- per-lane weights (neuron n = nt*16 + l16) ---
    float w1t[3][4], w1S[4], w1V[3][4], b1v[3][4];  // S-term only exists for mlp 0
    float wo[3][4], b2v[3][4];
    {
        const MlpW mlps[3] = { mS, mDr, mDv };
        #pragma unroll
        for (int m = 0; m < 3; ++m) {
            #pragma unroll
            for (int nt = 0; nt < 4; ++nt) {
                int n = nt * 16 + l16;
                bool ok = (n < 50);
                const float* W1 = mlps[m].W1;
                if (m == 0) {
                    w1t[m][nt] = ok ? W1[n * 3 + 0] : 0.f;
                    w1S[nt]    = ok ? W1[n * 3 + 1] : 0.f;
                    w1V[m][nt] = ok ? W1[n * 3 + 2] : 0.f;
                } else {
                    w1t[m][nt] = ok ? W1[n * 2 + 0] : 0.f;
                    w1V[m][nt] = ok ? W1[n * 2 + 1] : 0.f;
                }
                b1v[m][nt] = ok ? mlps[m].b1[n] : 0.f;
                wo[m][nt]  = ok ? mlps[m].Wo[n] : 0.f;
                b2v[m][nt] = ok ? mlps[m].b2[n] : 0.f;
            }
        }
    }

    const float rate = ratep[0];
    const float sqh  = sqrtf(H_STEP);
    float S[8], V[8];
    {
        float s0 = S0p[0], v0 = V0p[0];
        #pragma unroll
        for (int r = 0; r < 8; ++r) { S[r] = s0; V[r] = v0; }
    }
    const int pbase = blockIdx.x * PATHS_PER_BLOCK + wv * 16 + hi * 8;
    float* myH = sH[wv];

    for (int step = 0; step < N_STEPS; ++step) {
        const float t   = step * H_STEP;
        const float rho = rhoTab[step];
        const float sq  = sqTab[step];

        float dW[8], dB[8];
        #pragma unroll
        for (int r = 0; r < 8; ++r) {
            int p = pbase + r;                       // broadcast load per half
            dW[r] = sqh * z [p * N_STEPS + step];
            dB[r] = sqh * z1[p * N_STEPS + step];
        }

        float outv[3][8];
        #pragma unroll
        for (int m = 0; m < 3; ++m) {
            // ---- layer 1 (VALU, K<=3 + bias), write h1 in [M][K] tile ----
            #pragma unroll
            for (int nt = 0; nt < 4; ++nt) {
                #pragma unroll
                for (int r = 0; r < 8; ++r) {
                    float hv;
                    if (m == 0) {  // compile-time branch: 3-input MLP
                        hv = fmaf(w1t[m][nt], t,
                             fmaf(w1S[nt], S[r],
                             fmaf(w1V[m][nt], V[r], b1v[m][nt])));
                    } else {       // 2-input MLPs: no S term (avoid dead 0*x FMAs)
                        hv = fmaf(w1t[m][nt], t,
                             fmaf(w1V[m][nt], V[r], b1v[m][nt]));
                    }
                    hv = fmaxf(hv, 0.f);
                    myH[(r + 8 * hi) * H1_STRIDE + nt * 16 + l16] = hv;
                }
            }
            // ---- layer 2: 16x16x4 f32 WMMA, bias folded into C ----
            v8f acc[4];
            #pragma unroll
            for (int nt = 0; nt < 4; ++nt) {
                float b = b2v[m][nt];
                acc[nt] = (v8f){ b, b, b, b, b, b, b, b };
            }
            #pragma unroll
            for (int kc = 0; kc < NKC; ++kc) {
                v2f a = *(const v2f*)&myH[l16 * H1_STRIDE + 4 * kc + 2 * hi];
                #pragma unroll
                for (int nt = 0; nt < 4; ++nt) {
                    v2f b = sB[((m * NKC + kc) * NNT + nt) * 32 + lane];
                    acc[nt] = __builtin_amdgcn_wmma_f32_16x16x4_f32(
                        false, a, false, b, (short)0, acc[nt], false, false);
                }
            }
            // ---- ReLU + Wo dot + 16-lane butterfly reduce -> broadcast layout ----
            #pragma unroll
            for (int r = 0; r < 8; ++r) {
                float s = 0.f;
                #pragma unroll
                for (int nt = 0; nt < 4; ++nt)
                    s = fmaf(fmaxf(acc[nt][r], 0.f), wo[m][nt], s);
                #pragma unroll
                for (int off = 1; off < 16; off <<= 1)
                    s += __shfl_xor(s, off, 16);
                outv[m][r] = s;
            }
        }

        // ---- SDE update (all MLPs consumed old S,V) ----
        #pragma unroll
        for (int r = 0; r < 8; ++r) {
            float diffS = softplusf(outv[0][r]);
            float Sn = fmaf(S[r], 1.f + rate * H_STEP, diffS * dW[r]);
            S[r] = fmaxf(Sn, 0.f);
            float dv = softplusf(outv[2][r]);
            float Vn = V[r] + outv[1][r] * H_STEP + dv * (sq * dB[r] + rho * dW[r]);
            V[r] = fmaxf(Vn, 0.f);
        }

        // ---- payoff accumulation on even (1-based) steps ----
        if (step & 1) {
            int row = (step - 1) >> 1;     // 0..23
            #pragma unroll
            for (int c = 0; c < 3; ++c) {
                int tgt = c * 16 + l16;    // [0,40): 0..19 call, 20..39 put
                if (tgt < 40) {
                    int k = tgt % 20;
                    float strike = (k < 10) ? (55.f + 5.f * k) : (50.f + 5.f * k);
                    bool isCall = (tgt < 20);
                    float ssum = 0.f;
                    #pragma unroll
                    for (int r = 0; r < 8; ++r) {
                        float d = S[r] - strike;
                        ssum += isCall ? fmaxf(d, 0.f) : fmaxf(-d, 0.f);
                    }
                    atomicAdd(&sAcc[row * 40 + tgt], ssum);
                }
            }
        }
    }

    __syncthreads();
    for (int i = tid; i < 24 * 40; i += BLOCK_T) atomicAdd(&gAcc[i], sAcc[i]);
}

// ---------------- finalize: mean * discount, reference concat order --------
__global__ void fin_kernel(const float* __restrict__ gAcc,
                           const float* __restrict__ ratep,
                           float* __restrict__ out) {
    int o = blockIdx.x * blockDim.x + threadIdx.x;
    if (o >= 960) return;
    int rr = o / 10, c = o % 10;
    int j = rr % 24, seg = rr / 24;
    int idx = (seg == 0) ? (10 + c)       // call, strikes 10..19
            : (seg == 1) ? (20 + c)       // put,  strikes 0..9
            : (seg == 2) ? (c)            // call, strikes 0..9
                         : (30 + c);      // put,  strikes 10..19
    float t = (2 * j + 1) * H_STEP;       // TIMEGRID[indices-1]
    float disc = expf(-ratep[0] * t);
    out[o] = gAcc[j * 40 + idx] * disc * (1.0f / MC_PATHS);
}

// ---------------- host launcher --------------------------------------------
extern "C" void kernel_launch(void* const* d_in, const int* in_sizes, int n_in,
                              void* d_out, int out_size, void* d_ws, size_t ws_size,
                              hipStream_t stream) {
    MlpW m[4];
    const float *S0, *V0, *rate, *z, *z1;

    if (n_in >= 26) {
        // Leaf-flattened pytrees: 4 MLPs x 5 leaves, then S0,V0,rate,indices,z,z1.
        // Detect leaf order inside each MLP dict via p_diff.W1 size (150) vs Wo (50).
        bool insertion = (in_sizes[0] > 100);   // insertion: Ws0,Ws1,bs0,bs1,Wo
        for (int g = 0; g < 4; ++g) {
            const float* p[5];
            for (int k = 0; k < 5; ++k) p[k] = (const float*)d_in[g * 5 + k];
            if (insertion) m[g] = { p[0], p[1], p[2], p[3], p[4] };
            else           m[g] = { p[1], p[2], p[3], p[4], p[0] };  // sorted: Wo,Ws0,Ws1,bs0,bs1
        }
        S0   = (const float*)d_in[20];
        V0   = (const float*)d_in[21];
        rate = (const float*)d_in[22];
        z    = (const float*)d_in[24];
        z1   = (const float*)d_in[25];
    } else {
        // One concatenated flat buffer per top-level param (insertion order W1,W2,b1,b2,Wo).
        const int din[4] = { 3, 2, 2, 1 };
        const int wid[4] = { 50, 50, 50, 5 };
        for (int g = 0; g < 4; ++g) {
            const float* base = (const float*)d_in[g];
            const float* W1 = base;
            const float* W2 = W1 + wid[g] * din[g];
            const float* b1 = W2 + wid[g] * wid[g];
            const float* b2 = b1 + wid[g];
            const float* Wo = b2 + wid[g];
            m[g] = { W1, W2, b1, b2, Wo };
        }
        S0   = (const float*)d_in[4];
        V0   = (const float*)d_in[5];
        rate = (const float*)d_in[6];
        z    = (const float*)d_in[8];
        z1   = (const float*)d_in[9];
    }

    float* wsf    = (float*)d_ws;
    float* gAcc   = wsf;            // 960
    float* rhoTab = wsf + 960;      // 48
    float* sqTab  = wsf + 1008;     // 48

    // d_in order: 0=p_diff, 1=p_diffV, 2=p_driftV, 3=p_rho
    prep_kernel<<<1, 1024, 0, stream>>>(m[3], gAcc, rhoTab, sqTab);
    sde_kernel<<<MC_PATHS / PATHS_PER_BLOCK, BLOCK_T, 0, stream>>>(
        m[0], /*driftV*/ m[2], /*diffV*/ m[1],
        S0, V0, rate, z, z1, rhoTab, sqTab, gAcc);
    fin_kernel<<<(960 + 255) / 256, 256, 0, stream>>>(gAcc, rate, (float*)d_out);
}